// SparseSelfAttention_88278757802165
// MI455X (gfx1250) — compile-verified
//
#include <hip/hip_runtime.h>

#define HIDDEN 1024
#define SEQ    4096
#define NH     16
#define HD     64
#define WIN    512
#define HALF   256

typedef float v2f __attribute__((ext_vector_type(2)));
typedef float v8f __attribute__((ext_vector_type(8)));

__device__ __forceinline__ v8f wmma4(v2f a, v2f b, v8f c) {
  // D = A(16x4 f32) * B(4x16 f32) + C(16x16 f32)
  return __builtin_amdgcn_wmma_f32_16x16x4_f32(false, a, false, b, (short)0, c,
                                               false, false);
}

// CDNA5 async global->LDS copy (ASYNCcnt-tracked, bypasses VGPRs).
// Per active lane: copies 16 bytes from the global address in g to the LDS address in ldsOff.
__device__ __forceinline__ void async_copy_b128(unsigned ldsOff, const float* g) {
  asm volatile("global_load_async_to_lds_b128 %0, %1, off"
               :: "v"(ldsOff), "v"(g)
               : "memory");
}
__device__ __forceinline__ unsigned lds_off(const void* p) {
  // flat LDS address = {shared aperture, 32-bit LDS offset}
  return (unsigned)(unsigned long long)p;
}
__device__ __forceinline__ void wait_async0() {
#if __has_builtin(__builtin_amdgcn_s_wait_asynccnt)
  __builtin_amdgcn_s_wait_asynccnt(0);
#else
  asm volatile("s_wait_asynccnt 0x0" ::: "memory");
#endif
}
__device__ __forceinline__ void wait_async3() {
#if __has_builtin(__builtin_amdgcn_s_wait_asynccnt)
  __builtin_amdgcn_s_wait_asynccnt(3);
#else
  asm volatile("s_wait_asynccnt 0x3" ::: "memory");
#endif
}

// ---------------------------------------------------------------------------
// GEMM: out[s,n] = sum_k X[s,k]*W[k,n] + bias[n];  M=4096, N=1024, K=1024
// 256 threads (8 waves). Block tile M=128 x N=64, K-step 16, double-buffered
// via async-to-LDS copies overlapping the WMMA stream.
// ---------------------------------------------------------------------------
#define XS_STRIDE 20   // mult of 4 (B128-aligned rows); 20m mod 64 distinct -> conflict-free
#define WS_STRIDE 80   // mult of 4; half-wave rows land on disjoint banks

__global__ __launch_bounds__(256)
void gemm_f32_kernel(const float* __restrict__ X, const float* __restrict__ W,
                     const float* __restrict__ bias, float* __restrict__ out) {
  __shared__ float Xs[2][128 * XS_STRIDE];
  __shared__ float Ws[2][16 * WS_STRIDE];

  const int tid  = threadIdx.x;
  const int wave = tid >> 5;
  const int lane = tid & 31;
  const int l15  = lane & 15;
  const int lh   = lane >> 4;      // 0: K pair {0,1}, 1: K pair {2,3}
  const int nb0  = blockIdx.x * 64;
  const int mb0  = blockIdx.y * 128;

  // per-thread staging coordinates (3 async B128 per thread per K-step)
  const int xr  = tid >> 1;
  const int xc8 = (tid & 1) * 8;
  const int wr  = tid >> 4;
  const int wc4 = (tid & 15) * 4;
  const float* gX = &X[(size_t)(mb0 + xr) * HIDDEN + xc8];
  const float* gW = &W[(size_t)wr * HIDDEN + nb0 + wc4];

  v8f acc[4];
#pragma unroll
  for (int i = 0; i < 4; ++i) acc[i] = (v8f){0.f,0.f,0.f,0.f,0.f,0.f,0.f,0.f};

  // prologue: stage tile 0 into buffer 0
  async_copy_b128(lds_off(&Xs[0][xr * XS_STRIDE + xc8]),     gX);
  async_copy_b128(lds_off(&Xs[0][xr * XS_STRIDE + xc8 + 4]), gX + 4);
  async_copy_b128(lds_off(&Ws[0][wr * WS_STRIDE + wc4]),     gW);

  for (int kt = 0; kt < 64; ++kt) {
    const int cur = kt & 1;
    if (kt + 1 < 64) {   // issue next tile into the other buffer
      const int nxt = cur ^ 1;
      const float* nX = gX + (kt + 1) * 16;
      const float* nW = gW + (size_t)(kt + 1) * 16 * HIDDEN;
      async_copy_b128(lds_off(&Xs[nxt][xr * XS_STRIDE + xc8]),     nX);
      async_copy_b128(lds_off(&Xs[nxt][xr * XS_STRIDE + xc8 + 4]), nX + 4);
      async_copy_b128(lds_off(&Ws[nxt][wr * WS_STRIDE + wc4]),     nW);
      wait_async3();     // retire current buffer's 3 copies (in-order completion)
    } else {
      wait_async0();
    }
    __syncthreads();     // current buffer visible to all waves

    const float* xs = Xs[cur];
    const float* ws = Ws[cur];
#pragma unroll
    for (int kk = 0; kk < 4; ++kk) {
      const int ko = kk * 4 + 2 * lh;
      v2f a = *(const v2f*)&xs[(wave * 16 + l15) * XS_STRIDE + ko];
#pragma unroll
      for (int nt = 0; nt < 4; ++nt) {
        v2f b;
        b.x = ws[ko * WS_STRIDE + nt * 16 + l15];
        b.y = ws[(ko + 1) * WS_STRIDE + nt * 16 + l15];
        acc[nt] = wmma4(a, b, acc[nt]);
      }
    }
    __syncthreads();     // all waves done with cur before it is overwritten
  }

#pragma unroll
  for (int nt = 0; nt < 4; ++nt) {
    const int col = nb0 + nt * 16 + l15;
    const float bv = bias[col];
#pragma unroll
    for (int i = 0; i < 8; ++i) {
      const int row = mb0 + wave * 16 + i + 8 * lh;   // C layout: vgpr i -> M=i / M=i+8
      out[(size_t)row * HIDDEN + col] = acc[nt][i] + bv;
    }
  }
}

// ---------------------------------------------------------------------------
// Banded attention. One block per (strip of 64 query rows, chunk, head).
//  Phase 1: async-copy Q strip (64x64) into LDS (scale folded into scatter).
//  Phase 2: full-score 16x16 WMMA tiles over 9 j-blocks of 64 keys,
//           scatter in-band elements into Lband[64][512] (stride 516).
//           Zero-padded OOB keys give exact score 0 (as in reference).
//  Phase 3: softmax per row (wave32 shfl reductions); write probs (global+LDS).
//  Phase 4: ctx = probs_full @ V via WMMA, diagonal band reads with zero fill.
// ---------------------------------------------------------------------------
#define BAND_STRIDE 516   // 515 ≡ 3 mod 64 -> conflict-free diagonal access
#define QK_STRIDE   68    // mult of 4 (B128-aligned), 68m mod 64 distinct

__global__ __launch_bounds__(256)
void attn_kernel(const float* __restrict__ qb, const float* __restrict__ kb,
                 const float* __restrict__ vb, float* __restrict__ probs,
                 float* __restrict__ ctx) {
  extern __shared__ float smem[];
  float* Lband = smem;                               // 64 x 516
  float* Qs    = smem + 64 * BAND_STRIDE;            // 64 x 68
  float* KVs   = Qs + 64 * QK_STRIDE;                // 64 x 68 (K then V)

  const int strip = blockIdx.x;    // 0..7
  const int c     = blockIdx.y;    // 0..7
  const int h     = blockIdx.z;    // 0..15
  const int tid   = threadIdx.x;
  const int wave  = tid >> 5;
  const int lane  = tid & 31;
  const int l15   = lane & 15;
  const int lh    = lane >> 4;
  const int r0    = strip * 64;        // chunk-relative strip base
  const int qrow0 = c * WIN + r0;      // global query row base

  // Phase 1: Q strip (async copy, unscaled)
  for (int idx = tid; idx < 64 * 16; idx += 256) {
    const int i = idx >> 4, d4 = (idx & 15) * 4;
    async_copy_b128(lds_off(&Qs[i * QK_STRIDE + d4]),
                    &qb[(size_t)(qrow0 + i) * HIDDEN + h * 64 + d4]);
  }
  wait_async0();
  __syncthreads();

  // Phase 2: scores into band
  for (int jb = 0; jb < 9; ++jb) {
    for (int idx = tid; idx < 64 * 16; idx += 256) {   // stage 64 key rows
      const int i = idx >> 4, d4 = (idx & 15) * 4;
      const int pos = c * WIN + r0 + jb * 64 + i - HALF;  // actual key position
      if ((unsigned)pos < (unsigned)SEQ) {
        async_copy_b128(lds_off(&KVs[i * QK_STRIDE + d4]),
                        &kb[(size_t)pos * HIDDEN + h * 64 + d4]);
      } else {
        float4 z = {0.f, 0.f, 0.f, 0.f};
        *(float4*)&KVs[i * QK_STRIDE + d4] = z;          // zero padding
      }
    }
    wait_async0();
    __syncthreads();
    for (int t = wave; t < 16; t += 8) {       // 2 tiles per wave, wave-uniform
      const int mt = t >> 2, jt = t & 3;
      const int jlo = jb * 64 + jt * 16;       // strip-relative col base
      const int wmin = jlo - (mt * 16 + 15);
      const int wmax = jlo + 15 - mt * 16;
      if (wmax >= 0 && wmin < WIN) {
        v8f s = (v8f){0.f,0.f,0.f,0.f,0.f,0.f,0.f,0.f};
#pragma unroll
        for (int kk = 0; kk < 16; ++kk) {      // D=64, K-step 4
          const int ko = kk * 4 + 2 * lh;
          v2f a = *(const v2f*)&Qs[(mt * 16 + l15) * QK_STRIDE + ko];
          v2f b = *(const v2f*)&KVs[(jt * 16 + l15) * QK_STRIDE + ko];
          s = wmma4(a, b, s);
        }
#pragma unroll
        for (int i = 0; i < 8; ++i) {
          const int rs = mt * 16 + i + 8 * lh;
          const int w  = jlo + l15 - rs;
          if ((unsigned)w < (unsigned)WIN)
            Lband[rs * BAND_STRIDE + w] = 0.125f * s[i];   // 1/sqrt(D)
        }
      }
    }
    __syncthreads();
  }

  // Phase 3: softmax per row (wave handles 8 rows)
  for (int rr = 0; rr < 8; ++rr) {
    const int rs = wave * 8 + rr;
    float* row = &Lband[rs * BAND_STRIDE];
    float vals[16];
    float mx = -3.4e38f;
#pragma unroll
    for (int i = 0; i < 16; ++i) { vals[i] = row[lane + 32 * i]; mx = fmaxf(mx, vals[i]); }
#pragma unroll
    for (int off = 16; off > 0; off >>= 1) mx = fmaxf(mx, __shfl_xor(mx, off, 32));
    float sum = 0.f;
#pragma unroll
    for (int i = 0; i < 16; ++i) { vals[i] = __expf(vals[i] - mx); sum += vals[i]; }
#pragma unroll
    for (int off = 16; off > 0; off >>= 1) sum += __shfl_xor(sum, off, 32);
    const float inv = 1.0f / sum;
    const size_t gro = (size_t)(h * SEQ + qrow0 + rs) * WIN;
#pragma unroll
    for (int i = 0; i < 16; ++i) {
      const float p = vals[i] * inv;
      row[lane + 32 * i] = p;
      probs[gro + lane + 32 * i] = p;
    }
  }
  __syncthreads();

  // Phase 4: ctx = probs_full @ V. Each wave owns tiles (mt, nt0) and (mt, nt0+1).
  const int mt  = wave >> 1;
  const int nt0 = (wave & 1) * 2;
  v8f c0 = (v8f){0.f,0.f,0.f,0.f,0.f,0.f,0.f,0.f};
  v8f c1 = c0;
  for (int jb = 0; jb < 9; ++jb) {
    for (int idx = tid; idx < 64 * 16; idx += 256) {   // stage 64 value rows
      const int i = idx >> 4, d4 = (idx & 15) * 4;
      const int pos = c * WIN + r0 + jb * 64 + i - HALF;
      if ((unsigned)pos < (unsigned)SEQ) {
        async_copy_b128(lds_off(&KVs[i * QK_STRIDE + d4]),
                        &vb[(size_t)pos * HIDDEN + h * 64 + d4]);
      } else {
        float4 z = {0.f, 0.f, 0.f, 0.f};
        *(float4*)&KVs[i * QK_STRIDE + d4] = z;
      }
    }
    wait_async0();
    __syncthreads();
#pragma unroll
    for (int kk = 0; kk < 16; ++kk) {
      const int jl = jb * 64 + kk * 4 + 2 * lh;  // strip-relative key index (a.x)
      const int rs = mt * 16 + l15;
      const int w0 = jl - rs;
      v2f a;
      a.x = ((unsigned)w0       < (unsigned)WIN) ? Lband[rs * BAND_STRIDE + w0]     : 0.f;
      a.y = ((unsigned)(w0 + 1) < (unsigned)WIN) ? Lband[rs * BAND_STRIDE + w0 + 1] : 0.f;
      const int vr = (kk * 4 + 2 * lh) * QK_STRIDE;
      v2f b0, b1;
      b0.x = KVs[vr + nt0 * 16 + l15];
      b0.y = KVs[vr + QK_STRIDE + nt0 * 16 + l15];
      b1.x = KVs[vr + (nt0 + 1) * 16 + l15];
      b1.y = KVs[vr + QK_STRIDE + (nt0 + 1) * 16 + l15];
      c0 = wmma4(a, b0, c0);
      c1 = wmma4(a, b1, c1);
    }
    __syncthreads();
  }
#pragma unroll
  for (int i = 0; i < 8; ++i) {
    const int rs = mt * 16 + i + 8 * lh;
    ctx[(size_t)(qrow0 + rs) * HIDDEN + h * 64 + nt0 * 16 + l15]       = c0[i];
    ctx[(size_t)(qrow0 + rs) * HIDDEN + h * 64 + (nt0 + 1) * 16 + l15] = c1[i];
  }
}

// ---------------------------------------------------------------------------
extern "C" void kernel_launch(void* const* d_in, const int* in_sizes, int n_in,
                              void* d_out, int out_size, void* d_ws, size_t ws_size,
                              hipStream_t stream) {
  const float* hs = (const float*)d_in[0];
  const float* Wq = (const float*)d_in[1];
  const float* bq = (const float*)d_in[2];
  const float* Wk = (const float*)d_in[3];
  const float* bk = (const float*)d_in[4];
  const float* Wv = (const float*)d_in[5];
  const float* bv = (const float*)d_in[6];
  const float* Wo = (const float*)d_in[7];
  const float* bo = (const float*)d_in[8];

  float* out   = (float*)d_out;                       // [4096,1024]
  float* probs = out + (size_t)SEQ * HIDDEN;          // [16,4096,512]

  float* qbuf = (float*)d_ws;
  float* kbuf = qbuf + (size_t)SEQ * HIDDEN;
  float* vbuf = kbuf + (size_t)SEQ * HIDDEN;
  float* ctxb = vbuf + (size_t)SEQ * HIDDEN;

  const dim3 gg(HIDDEN / 64, SEQ / 128, 1);
  gemm_f32_kernel<<<gg, 256, 0, stream>>>(hs, Wq, bq, qbuf);
  gemm_f32_kernel<<<gg, 256, 0, stream>>>(hs, Wk, bk, kbuf);
  gemm_f32_kernel<<<gg, 256, 0, stream>>>(hs, Wv, bv, vbuf);

  const size_t smem = (size_t)(64 * BAND_STRIDE + 2 * 64 * QK_STRIDE) * sizeof(float);
  attn_kernel<<<dim3(8, 8, NH), 256, smem, stream>>>(qbuf, kbuf, vbuf, probs, ctxb);

  gemm_f32_kernel<<<gg, 256, 0, stream>>>(ctxb, Wo, bo, out);
}